// TimeEncoder_31731218383102
// MI455X (gfx1250) — compile-verified
//
#include <hip/hip_runtime.h>

// Embedding gather: out[n] = concat(W0[T[n,0]], W1[T[n,1]], W2[T[n,2]], W3[T[n,3]])
// N rows, per-table embed dim 32, output row = 128 floats.
// Pure memory-bound gather: ~1.02 GB NT stores + 32 MB index reads at 23.3 TB/s.

typedef float v4f __attribute__((ext_vector_type(4)));
typedef int   v4i __attribute__((vector_size(16)));

// Pointer flavors for the gfx1250 async global->LDS DMA builtin:
//   arg0: global (AS1) pointer to int4, arg1: LDS (AS3) pointer to int4
typedef __attribute__((address_space(1))) v4i* gv4p;
typedef __attribute__((address_space(3))) v4i* lv4p;

#define TL0 366
#define TL1 24
#define TL2 7
#define TL3 60
// float4 (16B) counts per table: TLi * 32 floats / 4
#define C40 (TL0 * 8) // 2928
#define C41 (TL1 * 8) // 192
#define C42 (TL2 * 8) // 56
#define C43 (TL3 * 8) // 480
#define B40 0
#define B41 (B40 + C40) // 2928
#define B42 (B41 + C41) // 3120
#define B43 (B42 + C42) // 3176
#define TOT4 (B43 + C43) // 3656 float4 = 58,496 bytes of LDS

__launch_bounds__(256, 4)
__global__ void te_gather_kernel(const int* __restrict__ T,
                                 const float* __restrict__ W0,
                                 const float* __restrict__ W1,
                                 const float* __restrict__ W2,
                                 const float* __restrict__ W3,
                                 float* __restrict__ out,
                                 int N)
{
    __shared__ v4f lds4[TOT4];
    const int tid = threadIdx.x;

    // ---- Stage all four tables into LDS (gfx1250 async DMA, ASYNCcnt) ----
#if __has_builtin(__builtin_amdgcn_global_load_async_to_lds_b128)
    {
        lv4p l = (lv4p)lds4;
        for (int i = tid; i < C40; i += 256)
            __builtin_amdgcn_global_load_async_to_lds_b128((gv4p)W0 + i, l + (B40 + i), 0, 0);
        for (int i = tid; i < C41; i += 256)
            __builtin_amdgcn_global_load_async_to_lds_b128((gv4p)W1 + i, l + (B41 + i), 0, 0);
        for (int i = tid; i < C42; i += 256)
            __builtin_amdgcn_global_load_async_to_lds_b128((gv4p)W2 + i, l + (B42 + i), 0, 0);
        for (int i = tid; i < C43; i += 256)
            __builtin_amdgcn_global_load_async_to_lds_b128((gv4p)W3 + i, l + (B43 + i), 0, 0);
#if __has_builtin(__builtin_amdgcn_s_wait_asynccnt)
        __builtin_amdgcn_s_wait_asynccnt(0);
#else
        asm volatile("s_wait_asynccnt 0" ::: "memory");
#endif
    }
#else
    {
        const v4f* g0 = (const v4f*)W0;
        const v4f* g1 = (const v4f*)W1;
        const v4f* g2 = (const v4f*)W2;
        const v4f* g3 = (const v4f*)W3;
        for (int i = tid; i < C40; i += 256) lds4[B40 + i] = g0[i];
        for (int i = tid; i < C41; i += 256) lds4[B41 + i] = g1[i];
        for (int i = tid; i < C42; i += 256) lds4[B42 + i] = g2[i];
        for (int i = tid; i < C43; i += 256) lds4[B43 + i] = g3[i];
    }
#endif
    __syncthreads();

    // ---- Gather + stream out ----
    // One wave32 per output row: lane j -> dim d = j/8, float4 column j%8.
    const int lane = tid & 31;
    const int warp = tid >> 5;
    const int d    = lane >> 3;
    const int sub  = lane & 7;
    const int base4 = (d == 0) ? B40 : (d == 1) ? B41 : (d == 2) ? B42 : B43;

    v4f* out4 = (v4f*)out;
    const long long rstep = (long long)gridDim.x * 8;
    long long row = (long long)blockIdx.x * 8 + warp;

    // Unroll x4: keep 4 independent index loads / LDS gathers / NT stores in
    // flight per wave to saturate the HBM write stream.
    for (; row + 3 * rstep < (long long)N; row += 4 * rstep) {
        const long long r0 = row;
        const long long r1 = row + rstep;
        const long long r2 = row + 2 * rstep;
        const long long r3 = row + 3 * rstep;
        const int t0 = T[r0 * 4 + d];
        const int t1 = T[r1 * 4 + d];
        const int t2 = T[r2 * 4 + d];
        const int t3 = T[r3 * 4 + d];
        const v4f a = lds4[base4 + t0 * 8 + sub];
        const v4f b = lds4[base4 + t1 * 8 + sub];
        const v4f c = lds4[base4 + t2 * 8 + sub];
        const v4f e = lds4[base4 + t3 * 8 + sub];
        __builtin_nontemporal_store(a, &out4[r0 * 32 + lane]);
        __builtin_nontemporal_store(b, &out4[r1 * 32 + lane]);
        __builtin_nontemporal_store(c, &out4[r2 * 32 + lane]);
        __builtin_nontemporal_store(e, &out4[r3 * 32 + lane]);
    }
    // Remainder rows.
    for (; row < (long long)N; row += rstep) {
        const int t = T[row * 4 + d];
        const v4f v = lds4[base4 + t * 8 + sub];
        __builtin_nontemporal_store(v, &out4[row * 32 + lane]);
    }
}

extern "C" void kernel_launch(void* const* d_in, const int* in_sizes, int n_in,
                              void* d_out, int out_size, void* d_ws, size_t ws_size,
                              hipStream_t stream)
{
    (void)n_in; (void)d_ws; (void)ws_size; (void)out_size;
    const int*   T  = (const int*)d_in[0];
    const float* W0 = (const float*)d_in[1];
    const float* W1 = (const float*)d_in[2];
    const float* W2 = (const float*)d_in[3];
    const float* W3 = (const float*)d_in[4];
    float* out = (float*)d_out;

    const int N = in_sizes[0] / 4; // T is [N,4]

    int blocks = (N + 7) / 8;
    if (blocks > 4096) blocks = 4096; // persistent-ish: amortize 57 KB LDS fill
    if (blocks < 1) blocks = 1;

    te_gather_kernel<<<blocks, 256, 0, stream>>>(T, W0, W1, W2, W3, out, N);
}